// QParameter_28080496181599
// MI455X (gfx1250) — compile-verified
//
#include <hip/hip_runtime.h>

// QParameter dequant: out[r, 2j]   = s[r,g]*((qw[r,j]>>4) - z[r,g]) + outliers[r,2j]
//                     out[r, 2j+1] = s[r,g]*((qw[r,j]&15) - z[r,g]) + outliers[r,2j+1]
// Shapes: qweight (4096,5504) int32 (one packed byte per int32), scale/zero (4096,86),
// outliers/out (4096,11008) fp32, groupsize 128.
//
// Roofline: ~451 MB single-use traffic -> ~19.4 us floor at 23.3 TB/s. Pure
// streaming, zero reuse => WMMA/TDM cannot reduce bytes; the gfx1250 features
// that matter are wave32 sizing, b128 vector ops, and NT temporal hints
// (working set >> 192 MB L2, every byte touched once).
//
// 16 outputs per thread: 2x b128 qweight loads + 4x b128 outlier loads +
// 4x b128 NT stores, one scale/zero pair (16 | groupsize). Exact grid:
// 4096 rows * 688 threads/row = 2,818,048 = 11,008 blocks * 256 threads.

typedef __attribute__((ext_vector_type(4))) float v4f;
typedef __attribute__((ext_vector_type(4))) int   v4i;

#define OUT_DIM         4096
#define IN_DIM          11008
#define N_GROUPS        86
#define PACKED_PER_ROW  (IN_DIM / 2)     // 5504
#define ELEMS_PER_THR   16
#define THREADS_PER_ROW (IN_DIM / ELEMS_PER_THR)          // 688
#define TOTAL_THREADS   (OUT_DIM * THREADS_PER_ROW)       // 2,818,048

// Two packed int32 bytes -> four dequantized outputs (hi nibble first).
__device__ __forceinline__ v4f dq4(int qa, int qb, float s, float z, v4f o) {
    v4f r;
    r.x = fmaf(s, (float)((qa >> 4) & 15) - z, o.x);  // even column
    r.y = fmaf(s, (float)( qa       & 15) - z, o.y);  // odd column
    r.z = fmaf(s, (float)((qb >> 4) & 15) - z, o.z);
    r.w = fmaf(s, (float)( qb       & 15) - z, o.w);
    return r;
}

__global__ __launch_bounds__(256) void
dequant4_stream(const int*   __restrict__ qw,
                const float* __restrict__ scale,
                const float* __restrict__ zero,
                const float* __restrict__ outliers,
                float*       __restrict__ out)
{
    const unsigned tid = blockIdx.x * blockDim.x + threadIdx.x;
    if (tid >= TOTAL_THREADS) return;

    const unsigned row = tid / THREADS_PER_ROW;        // const-divisor -> mul-shift
    const unsigned t   = tid - row * THREADS_PER_ROW;  // 0..687 within row
    const unsigned g   = t >> 3;                       // (t*16)/128: group index

    // scale/zero shared by 8 consecutive threads -> regular (cacheable) loads
    const float s = scale[row * N_GROUPS + g];
    const float z = zero [row * N_GROUPS + g];

    // 8 packed int32 = 8 bytes of nibbles = 16 outputs; two b128 NT loads
    const v4i* qp = (const v4i*)(qw + (size_t)row * PACKED_PER_ROW) + 2u * t;
    const v4i q0 = __builtin_nontemporal_load(qp);
    const v4i q1 = __builtin_nontemporal_load(qp + 1);

    const v4f* op = (const v4f*)(outliers + (size_t)row * IN_DIM) + 4u * t;
    const v4f o0 = __builtin_nontemporal_load(op);
    const v4f o1 = __builtin_nontemporal_load(op + 1);
    const v4f o2 = __builtin_nontemporal_load(op + 2);
    const v4f o3 = __builtin_nontemporal_load(op + 3);

    const v4f r0 = dq4(q0.x, q0.y, s, z, o0);
    const v4f r1 = dq4(q0.z, q0.w, s, z, o1);
    const v4f r2 = dq4(q1.x, q1.y, s, z, o2);
    const v4f r3 = dq4(q1.z, q1.w, s, z, o3);

    // single-use output stream -> NT stores (avoid L2 write-rinse churn)
    v4f* po = (v4f*)(out + (size_t)row * IN_DIM) + 4u * t;
    __builtin_nontemporal_store(r0, po);
    __builtin_nontemporal_store(r1, po + 1);
    __builtin_nontemporal_store(r2, po + 2);
    __builtin_nontemporal_store(r3, po + 3);
}

extern "C" void kernel_launch(void* const* d_in, const int* in_sizes, int n_in,
                              void* d_out, int out_size, void* d_ws, size_t ws_size,
                              hipStream_t stream) {
    (void)in_sizes; (void)n_in; (void)d_ws; (void)ws_size; (void)out_size;
    const int*   qweight  = (const int*)  d_in[0];
    const float* scale    = (const float*)d_in[1];
    const float* zero     = (const float*)d_in[2];
    const float* outliers = (const float*)d_in[3];
    // d_in[4] = bits (4), d_in[5] = groupsize (128): compile-time constants here
    float* out = (float*)d_out;

    const int block = 256;                                  // 8 wave32 waves
    const int grid  = (TOTAL_THREADS + block - 1) / block;  // 11,008, exact
    dequant4_stream<<<grid, block, 0, stream>>>(qweight, scale, zero, outliers, out);
}